// _ScaledDotProductAttention_66443144069307
// MI455X (gfx1250) — compile-verified
//
#include <hip/hip_runtime.h>
#include <math.h>

typedef __attribute__((ext_vector_type(2))) float v2f;
typedef __attribute__((ext_vector_type(4))) float v4f;
typedef __attribute__((ext_vector_type(8))) float v8f;

#define BSZ 4
#define NH 16
#define SEQ 1024
#define HD 64
#define ROWS 16          // query rows per workgroup (one WMMA M-tile)
#define SROW 1028        // padded LDS row stride (floats): 4112B = 257*16 -> v4f aligned, banks spread
#define NTHREADS 128     // 4 waves (wave32)

__global__ __launch_bounds__(NTHREADS)
void attn_fused_f32wmma(const float* __restrict__ q,
                        const float* __restrict__ kmat,
                        const float* __restrict__ v,
                        const float* __restrict__ prev,
                        const float* __restrict__ scale_p,
                        float* __restrict__ out,    // [BH, SEQ, HD]
                        float* __restrict__ outw,   // [BH, SEQ, SEQ]
                        float* __restrict__ outs)   // [BH, SEQ, SEQ]
{
    extern __shared__ float smem[];
    float* s_q   = smem;                   // ROWS*HD      = 1024 floats
    float* s_s   = s_q + ROWS * HD;        // ROWS*SROW    = 16448 floats
    float* s_red = s_s + ROWS * SROW;      // ROWS*8
    float* s_inv = s_red + ROWS * 8;       // ROWS

    const int tid  = threadIdx.x;
    const int wave = tid >> 5;
    const int lane = tid & 31;
    const int lm   = lane & 15;            // N (or M) index within a 16-wide half
    const int lh   = (lane >> 4) & 1;      // which half of the wave

    const int mt    = blockIdx.x & (SEQ / ROWS - 1);
    const int bh    = blockIdx.x / (SEQ / ROWS);
    const int mBase = mt * ROWS;

    const float scale = scale_p[0];

    const float* Q = q    + (size_t)bh * SEQ * HD;
    const float* K = kmat + (size_t)bh * HD * SEQ;   // pre-transposed: [HD, SEQ]
    const float* V = v    + (size_t)bh * SEQ * HD;
    const float* P = prev + (size_t)bh * SEQ * SEQ + (size_t)mBase * SEQ;
    float* OutO = out  + (size_t)bh * SEQ * HD  + (size_t)mBase * HD;
    float* OutW = outw + (size_t)bh * SEQ * SEQ + (size_t)mBase * SEQ;
    float* OutS = outs + (size_t)bh * SEQ * SEQ + (size_t)mBase * SEQ;

    // ---- Load Q tile (16 x 64) into LDS, coalesced ----
    for (int i = tid; i < ROWS * HD; i += NTHREADS)
        s_q[i] = Q[(size_t)(i >> 6) * HD + (i & (HD - 1))];
    __syncthreads();

    // ---- Phase 1: raw S-acc = Q K^T (per-wave: 16 tiles of 16x16, K=64 in steps of 4) ----
    for (int nt = 0; nt < 16; ++nt) {
        const int nBase = wave * 256 + nt * 16;
        v8f acc = {};
#pragma unroll
        for (int k0 = 0; k0 < HD; k0 += 4) {
            const int ka = k0 + 2 * lh;    // 16x16x4 f32 layout: v0 holds K=0|2, v1 holds K=1|3
            v2f a, b;
            a.x = s_q[lm * HD + ka + 0];
            a.y = s_q[lm * HD + ka + 1];
            b.x = K[(size_t)(ka + 0) * SEQ + nBase + lm];
            b.y = K[(size_t)(ka + 1) * SEQ + nBase + lm];
            acc = __builtin_amdgcn_wmma_f32_16x16x4_f32(
                false, a, false, b, (short)0, acc, false, false);
        }
#pragma unroll
        for (int r = 0; r < 8; ++r) {
            const int m = r + 8 * lh;      // C/D layout: VGPR r -> M=r (lanes 0-15), M=r+8 (16-31)
            s_s[m * SROW + nBase + lm] = acc[r];
        }
    }
    __syncthreads();

    // ---- Phase 1b: S = acc*scale + prev, streamed b128 + non-temporal -----------
    // prev is read-once, scores write-once: NT keeps L2 free for the reused K/V slices.
    for (int i4 = tid; i4 < (ROWS * SEQ) / 4; i4 += NTHREADS) {
        const int row = i4 >> 8;           // i4*4 >> 10
        const int col = (i4 & 255) * 4;
        const v4f p4 = __builtin_nontemporal_load((const v4f*)&P[(size_t)row * SEQ + col]);
        v4f s4 = *(const v4f*)&s_s[row * SROW + col];
        s4 = s4 * scale + p4;
        *(v4f*)&s_s[row * SROW + col] = s4;
        __builtin_nontemporal_store(s4, (v4f*)&OutS[(size_t)row * SEQ + col]);
    }
    __syncthreads();

    // ---- Phase 2: exact softmax over each of the 16 rows --------------
    {
        const int row = tid >> 3;          // 8 threads per row
        const int sub = tid & 7;
        float mx = -__builtin_inff();
        for (int c = sub; c < SEQ; c += 8)                 // stride-8: conflict-free banks
            mx = fmaxf(mx, s_s[row * SROW + c]);
        s_red[row * 8 + sub] = mx;
        __syncthreads();
        float rmax = s_red[row * 8];
#pragma unroll
        for (int i = 1; i < 8; ++i) rmax = fmaxf(rmax, s_red[row * 8 + i]);
        __syncthreads();
        float psum = 0.f;
        for (int c = sub; c < SEQ; c += 8) {
            const float e = __expf(s_s[row * SROW + c] - rmax);
            s_s[row * SROW + c] = e;
            psum += e;
        }
        s_red[row * 8 + sub] = psum;
        __syncthreads();
        if (sub == 0) {
            float rs = 0.f;
#pragma unroll
            for (int i = 0; i < 8; ++i) rs += s_red[row * 8 + i];
            s_inv[row] = 1.0f / rs;
        }
        __syncthreads();
    }
    // ---- Phase 2c: normalize + b128 non-temporal store of attention weights ----
    for (int i4 = tid; i4 < (ROWS * SEQ) / 4; i4 += NTHREADS) {
        const int row = i4 >> 8;
        const int col = (i4 & 255) * 4;
        v4f w4 = *(const v4f*)&s_s[row * SROW + col];
        w4 = w4 * s_inv[row];
        *(v4f*)&s_s[row * SROW + col] = w4;
        __builtin_nontemporal_store(w4, (v4f*)&OutW[(size_t)row * SEQ + col]);
    }
    __syncthreads();

    // ---- Phase 3: O = W V (16x64 tile, each wave one 16x16 N-tile, K=1024) ----
    {
        const int nBase = wave * 16;
        v8f acc = {};
        for (int k0 = 0; k0 < SEQ; k0 += 4) {
            const int ka = k0 + 2 * lh;
            v2f a, b;
            a.x = s_s[lm * SROW + ka + 0];   // bank = (4*lm + ka) % 64 -> conflict-free
            a.y = s_s[lm * SROW + ka + 1];
            b.x = V[(size_t)(ka + 0) * HD + nBase + lm];
            b.y = V[(size_t)(ka + 1) * HD + nBase + lm];
            acc = __builtin_amdgcn_wmma_f32_16x16x4_f32(
                false, a, false, b, (short)0, acc, false, false);
        }
#pragma unroll
        for (int r = 0; r < 8; ++r) {
            const int m = r + 8 * lh;
            OutO[(size_t)m * HD + nBase + lm] = acc[r];
        }
    }
}

extern "C" void kernel_launch(void* const* d_in, const int* in_sizes, int n_in,
                              void* d_out, int out_size, void* d_ws, size_t ws_size,
                              hipStream_t stream) {
    (void)in_sizes; (void)n_in; (void)out_size; (void)d_ws; (void)ws_size;
    const float* q     = (const float*)d_in[0];
    const float* k     = (const float*)d_in[1];
    const float* v     = (const float*)d_in[2];
    const float* prev  = (const float*)d_in[3];
    const float* scale = (const float*)d_in[4];

    float* out  = (float*)d_out;                                   // [4,16,1024,64]
    float* outw = out  + (size_t)BSZ * NH * SEQ * HD;              // [4,16,1024,1024]
    float* outs = outw + (size_t)BSZ * NH * SEQ * SEQ;             // [4,16,1024,1024]

    const size_t shmem =
        (size_t)(ROWS * HD + ROWS * SROW + ROWS * 8 + ROWS) * sizeof(float); // ~70.5 KB

    // Allow >64KB dynamic LDS (CDNA5: up to 320KB per workgroup)
    hipFuncSetAttribute((const void*)attn_fused_f32wmma,
                        hipFuncAttributeMaxDynamicSharedMemorySize, (int)shmem);

    const dim3 grid(BSZ * NH * (SEQ / ROWS));  // 4096 workgroups
    const dim3 block(NTHREADS);
    attn_fused_f32wmma<<<grid, block, shmem, stream>>>(q, k, v, prev, scale,
                                                       out, outw, outs);
}